// SelfAttention_53910429499722
// MI455X (gfx1250) — compile-verified
//
#include <hip/hip_runtime.h>
#include <hip/hip_bf16.h>

// Problem constants (from reference): B=4, C=64, H=W=24
#define HW    576            // 24*24
#define NCOL  2304           // B*HW
#define KDIM  576            // C*9 (im2col K for both convs)
#define MQKV  192            // 3*C
#define MOUT  64             // C

typedef __attribute__((ext_vector_type(2))) float v2f;
typedef __attribute__((ext_vector_type(8))) float v8f;

// ---------------------------------------------------------------------------
// 1) BatchNorm batch statistics -> fused per-channel scale/shift
//    scale[c] = gamma[c] * rsqrt(var+eps);  shift[c] = beta[c] - mean*scale[c]
// ---------------------------------------------------------------------------
__global__ void bn_stats_kernel(const float* __restrict__ x,
                                const float* __restrict__ gamma,
                                const float* __restrict__ beta,
                                float* __restrict__ scale,
                                float* __restrict__ shift) {
    const int c = blockIdx.x;           // 64 blocks
    __shared__ float s1[256], s2[256];
    const int tid = threadIdx.x;
    float sum = 0.f, sq = 0.f;
    for (int t = tid; t < 4 * HW; t += 256) {
        int b = t / HW, pos = t - b * HW;
        float v = x[(b * 64 + c) * HW + pos];
        sum += v; sq += v * v;
    }
    s1[tid] = sum; s2[tid] = sq; __syncthreads();
    for (int s = 128; s > 0; s >>= 1) {
        if (tid < s) { s1[tid] += s1[tid + s]; s2[tid] += s2[tid + s]; }
        __syncthreads();
    }
    if (tid == 0) {
        float mean = s1[0] * (1.f / 2304.f);
        float var  = s2[0] * (1.f / 2304.f) - mean * mean;
        float istd = rsqrtf(var + 1e-5f);
        float sc   = gamma[c] * istd;
        scale[c] = sc;
        shift[c] = beta[c] - mean * sc;
    }
}

// ---------------------------------------------------------------------------
// 2) im2col of the *normalized* input (padding -> 0, matching conv-after-BN)
//    Acol[kk][n], kk = i*9 + ky*3 + kx, n = b*576 + y*24 + x
// ---------------------------------------------------------------------------
__global__ void im2col_bn_kernel(const float* __restrict__ x,
                                 const float* __restrict__ scale,
                                 const float* __restrict__ shift,
                                 float* __restrict__ Acol) {
    int idx = blockIdx.x * blockDim.x + threadIdx.x;
    if (idx >= KDIM * NCOL) return;
    int kk = idx / NCOL, n = idx - kk * NCOL;         // n fast -> coalesced
    int i = kk / 9, r = kk - i * 9, ky = r / 3, kx = r - ky * 3;
    int b = n / HW, pos = n - b * HW, y = pos / 24, xx = pos - y * 24;
    int yy = y + ky - 1, xs = xx + kx - 1;
    float v = 0.f;
    if ((unsigned)yy < 24u && (unsigned)xs < 24u)
        v = x[(b * 64 + i) * HW + yy * 24 + xs] * scale[i] + shift[i];
    Acol[idx] = v;
}

// ---------------------------------------------------------------------------
// 3) im2col of out_v (stored as [c][n]) for the output conv, plain zero-pad
// ---------------------------------------------------------------------------
__global__ void im2col_plain_kernel(const float* __restrict__ ov,
                                    float* __restrict__ Acol) {
    int idx = blockIdx.x * blockDim.x + threadIdx.x;
    if (idx >= KDIM * NCOL) return;
    int kk = idx / NCOL, n = idx - kk * NCOL;
    int i = kk / 9, r = kk - i * 9, ky = r / 3, kx = r - ky * 3;
    int b = n / HW, pos = n - b * HW, y = pos / 24, xx = pos - y * 24;
    int yy = y + ky - 1, xs = xx + kx - 1;
    float v = 0.f;
    if ((unsigned)yy < 24u && (unsigned)xs < 24u)
        v = ov[i * NCOL + b * HW + yy * 24 + xs];
    Acol[idx] = v;
}

// ---------------------------------------------------------------------------
// 4) fp32 WMMA GEMM with 2x2 register blocking:
//    each wave computes a 32x32 output macro-tile = 4 accumulators.
//    Per K-step: 2 x b64 A-frag loads + 4 x b32 B loads feed 4 WMMAs
//    (1.5 VMEM per matrix op instead of 3 -> memory-issue no longer bounds).
//    If resid != nullptr: write D = acc + resid into NCHW output layout.
//    Grids are exact multiples of 4 waves -> EXEC all-1s for WMMA.
// ---------------------------------------------------------------------------
__global__ void gemm_wmma_f32_kernel(const float* __restrict__ A,
                                     const float* __restrict__ Bm,
                                     float* __restrict__ C,
                                     const float* __restrict__ resid,
                                     int Mtiles2) {             // M/32
    const int NT2 = NCOL / 32;                                  // 72 col tiles
    int wave = blockIdx.x * (blockDim.x >> 5) + (threadIdx.x >> 5);
    if (wave >= Mtiles2 * NT2) return;                          // uniform
    int mt = wave / NT2, nt = wave - mt * NT2;
    int lane = threadIdx.x & 31;
    int l15  = lane & 15;
    int hi   = lane >> 4;            // 0 | 1
    int koff = hi << 1;              // K sub-offset 0 | 2 (f32 frag layout)

    const float* Ap = A  + (mt * 32 + l15) * KDIM + koff;   // 8B aligned
    const float* Bp = Bm + koff * NCOL + nt * 32 + l15;

    v8f acc00 = {}, acc01 = {}, acc10 = {}, acc11 = {};
    #pragma unroll 2
    for (int k0 = 0; k0 < KDIM; k0 += 4) {
        v2f a0 = *(const v2f*)Ap;                   // rows mt*32 + l15
        v2f a1 = *(const v2f*)(Ap + 16 * KDIM);     // rows +16
        v2f b0, b1;
        b0.x = Bp[0];         b0.y = Bp[NCOL];      // cols nt*32 + l15
        b1.x = Bp[16];        b1.y = Bp[NCOL + 16]; // cols +16
        acc00 = __builtin_amdgcn_wmma_f32_16x16x4_f32(
                    false, a0, false, b0, (short)0, acc00, false, false);
        acc01 = __builtin_amdgcn_wmma_f32_16x16x4_f32(
                    false, a0, false, b1, (short)0, acc01, false, false);
        acc10 = __builtin_amdgcn_wmma_f32_16x16x4_f32(
                    false, a1, false, b0, (short)0, acc10, false, false);
        acc11 = __builtin_amdgcn_wmma_f32_16x16x4_f32(
                    false, a1, false, b1, (short)0, acc11, false, false);
        Ap += 4;
        Bp += 4 * NCOL;
    }

    v8f accs[4] = { acc00, acc01, acc10, acc11 };
    #pragma unroll
    for (int mi = 0; mi < 2; ++mi) {
        #pragma unroll
        for (int ni = 0; ni < 2; ++ni) {
            const v8f& acc = accs[mi * 2 + ni];
            int row0 = mt * 32 + mi * 16 + (hi << 3);  // lanes16-31 -> M+8
            int col  = nt * 32 + ni * 16 + l15;
            if (resid) {                    // output conv: NCHW + residual
                int b = col / HW, pos = col - b * HW;
                #pragma unroll
                for (int r = 0; r < 8; ++r) {
                    int oidx = (b * 64 + (row0 + r)) * HW + pos;
                    C[oidx] = acc[r] + resid[oidx];
                }
            } else {                        // qkv: row-major [M][2304]
                #pragma unroll
                for (int r = 0; r < 8; ++r)
                    C[(row0 + r) * NCOL + col] = acc[r];
            }
        }
    }
}

// ---------------------------------------------------------------------------
// 5) Per-(b,c) attention. logits are rank-1 (q_i * k_j * s) so the softmax
//    max is exactly a_i>=0 ? a_i*kmax : a_i*kmin. k/v live in LDS.
//    qkv layout: rows [0,64)=q, [64,128)=k, [128,192)=v, each row [b*576+pos].
// ---------------------------------------------------------------------------
__global__ void attn_kernel(const float* __restrict__ qkv,
                            float* __restrict__ ov) {
    const int bc = blockIdx.x;                   // 256 blocks = B*C
    const int b = bc >> 6, c = bc & 63;
    __shared__ float kb[HW], vb[HW];
    __shared__ float rmx[256], rmn[256];
    const int tid = threadIdx.x;

    const float* kp = qkv + (64 + c) * NCOL + b * HW;
    const float* vp = qkv + (128 + c) * NCOL + b * HW;
    for (int t = tid; t < HW; t += 256) { kb[t] = kp[t]; vb[t] = vp[t]; }
    __syncthreads();

    float kmx = -3.402823466e+38f, kmn = 3.402823466e+38f;
    for (int t = tid; t < HW; t += 256) {
        kmx = fmaxf(kmx, kb[t]); kmn = fminf(kmn, kb[t]);
    }
    rmx[tid] = kmx; rmn[tid] = kmn; __syncthreads();
    for (int s = 128; s > 0; s >>= 1) {
        if (tid < s) {
            rmx[tid] = fmaxf(rmx[tid], rmx[tid + s]);
            rmn[tid] = fminf(rmn[tid], rmn[tid + s]);
        }
        __syncthreads();
    }
    kmx = rmx[0]; kmn = rmn[0];

    const float sc = 0.125f;                     // 1/sqrt(C), C=64
    const float* qp = qkv + c * NCOL + b * HW;
    float* op = ov + c * NCOL + b * HW;
    for (int i = tid; i < HW; i += 256) {
        float a = qp[i] * sc;
        float m = (a >= 0.f) ? a * kmx : a * kmn;
        float s = 0.f, acc = 0.f;
        #pragma unroll 4
        for (int j = 0; j < HW; ++j) {
            float e = __expf(fmaf(a, kb[j], -m));  // v_exp_f32
            s   += e;
            acc  = fmaf(e, vb[j], acc);
        }
        op[i] = acc / s;
    }
}

// ---------------------------------------------------------------------------
// launch
// ---------------------------------------------------------------------------
extern "C" void kernel_launch(void* const* d_in, const int* in_sizes, int n_in,
                              void* d_out, int out_size, void* d_ws, size_t ws_size,
                              hipStream_t stream) {
    const float* x     = (const float*)d_in[0];   // [4,64,24,24]
    const float* gamma = (const float*)d_in[1];   // [64]
    const float* beta  = (const float*)d_in[2];   // [64]
    const float* wqkv  = (const float*)d_in[3];   // [192,64,3,3] = [192][576]
    const float* wout  = (const float*)d_in[4];   // [64,64,3,3]  = [64][576]
    float* out = (float*)d_out;                   // [4,64,24,24]

    float* ws    = (float*)d_ws;
    float* scale = ws;                            // 64
    float* shift = ws + 64;                       // 64
    float* Acol  = ws + 128;                      // 576*2304
    float* qkv   = Acol + (size_t)KDIM * NCOL;    // 192*2304
    float* ov    = qkv  + (size_t)MQKV * NCOL;    // 64*2304

    // 1) BN statistics -> fused scale/shift
    bn_stats_kernel<<<64, 256, 0, stream>>>(x, gamma, beta, scale, shift);

    // 2) normalized im2col panel
    const int tot = KDIM * NCOL;
    im2col_bn_kernel<<<(tot + 255) / 256, 256, 0, stream>>>(x, scale, shift, Acol);

    // 3) qkv = Wqkv * Acol   (M/32=6 x 72 macro-tiles, 4 waves/block, exact)
    gemm_wmma_f32_kernel<<<(6 * 72) / 4, 128, 0, stream>>>(
        wqkv, Acol, qkv, nullptr, 6);

    // 4) rank-1-logit softmax attention
    attn_kernel<<<256, 256, 0, stream>>>(qkv, ov);

    // 5) im2col of out_v
    im2col_plain_kernel<<<(tot + 255) / 256, 256, 0, stream>>>(ov, Acol);

    // 6) out = x + Wout * Acol   (M/32=2 x 72 macro-tiles, exact)
    gemm_wmma_f32_kernel<<<(2 * 72) / 4, 128, 0, stream>>>(
        wout, Acol, out, x, 2);
}